// SAGE_47699906789905
// MI455X (gfx1250) — compile-verified
//
#include <hip/hip_runtime.h>
#include <hip/hip_bf16.h>

typedef float v2f __attribute__((ext_vector_type(2)));
typedef float v8f __attribute__((ext_vector_type(8)));

// ---------------------------------------------------------------------------
// Zero a float buffer
// ---------------------------------------------------------------------------
__global__ void sage_zero_f32(float* __restrict__ p, long long n) {
    long long i = (long long)blockIdx.x * blockDim.x + threadIdx.x;
    if (i < n) p[i] = 0.0f;
}

// ---------------------------------------------------------------------------
// Edge aggregation: NBR[dst] += H[src], DEG[dst] += 1.
// One thread per (edge, float4). Feature table (25.6 MB) is L2-resident on
// MI455X (192 MB L2): the b128 gather and the f32 atomic scatter hit L2.
// ---------------------------------------------------------------------------
__global__ void sage_aggregate(const float4* __restrict__ H4,   // [N,16] float4
                               const int* __restrict__ esrc,
                               const int* __restrict__ edst,
                               float* __restrict__ NBR,
                               float* __restrict__ DEG,
                               long long n_items)               // n_edges * 16
{
    long long idx = (long long)blockIdx.x * blockDim.x + threadIdx.x;
    if (idx >= n_items) return;
    long long e = idx >> 4;
    int q = (int)(idx & 15);
    int s = esrc[e];
    int d = edst[e];
    float4 v = H4[(size_t)s * 16 + q];
    float* dst = &NBR[(size_t)d * 64 + q * 4];
    atomicAdd(dst + 0, v.x);
    atomicAdd(dst + 1, v.y);
    atomicAdd(dst + 2, v.z);
    atomicAdd(dst + 3, v.w);
    if (q == 0) atomicAdd(&DEG[d], 1.0f);
}

// ---------------------------------------------------------------------------
// Fused SAGE layer GEMM via V_WMMA_F32_16X16X4_F32 (full-precision f32 WMMA):
//   OUT[v] = relu?( X[v] @ Wself + (NBR[v]/max(DEG[v],1)) @ Wneigh + bias )
//
// Weights + bias are staged once per workgroup into LDS with the CDNA5 async
// global->LDS path (GLOBAL_LOAD_ASYNC_TO_LDS_B128, ASYNCcnt), then the WMMA
// K-loops read B fragments from LDS (ds_load) instead of re-fetching from
// global per step.
//
// Fragment layouts per CDNA5 ISA (wave32):
//   A 16x4 : lane l (l<16): {A[l][0],A[l][1]}  lane l>=16: {A[l-16][2],A[l-16][3]}
//   B 4x16 : lane l (l<16): {B[0][l],B[1][l]}  lane l>=16: {B[2][l-16],B[3][l-16]}
//   C/D    : vgpr g, lane<16 -> D[g][lane], lane>=16 -> D[g+8][lane-16]
// ---------------------------------------------------------------------------
#define GB_THREADS 256

__global__ void sage_gemm_wmma(const float* __restrict__ X,      // [N,64]
                               const float* __restrict__ NBR,    // [N,64] neighbor sums
                               const float* __restrict__ DEG,    // [N]
                               const float* __restrict__ Wself,  // [64,64] row-major
                               const float* __restrict__ Wneigh, // [64,64] row-major
                               const float* __restrict__ bias,   // [64]
                               float* __restrict__ OUT,          // [N,64]
                               int n_nodes, int do_relu)
{
    // LDS: Wself [0,4096) | Wneigh [4096,8192) | bias [8192,8256)
    __shared__ float sW[2 * 64 * 64 + 64];

    const int tid = threadIdx.x;

    // ---- async stage: 2 x 16KB weights (4 b128 chunks per thread each) ----
    const unsigned lds_base = (unsigned)(size_t)&sW[0];
    #pragma unroll
    for (int i = 0; i < 4; ++i) {
        const int chunk = tid + i * GB_THREADS;          // 0..1023
        unsigned lds0 = lds_base + chunk * 16;
        unsigned long long g0 =
            (unsigned long long)(const char*)Wself + (unsigned)(chunk * 16);
        asm volatile("global_load_async_to_lds_b128 %0, %1, off"
                     :: "v"(lds0), "v"(g0) : "memory");
        unsigned lds1 = lds_base + 16384u + chunk * 16;
        unsigned long long g1 =
            (unsigned long long)(const char*)Wneigh + (unsigned)(chunk * 16);
        asm volatile("global_load_async_to_lds_b128 %0, %1, off"
                     :: "v"(lds1), "v"(g1) : "memory");
    }
    if (tid < 16) {                                      // bias: 16 b128 chunks
        unsigned lds2 = lds_base + 32768u + tid * 16;
        unsigned long long g2 =
            (unsigned long long)(const char*)bias + (unsigned)(tid * 16);
        asm volatile("global_load_async_to_lds_b128 %0, %1, off"
                     :: "v"(lds2), "v"(g2) : "memory");
    }
    asm volatile("s_wait_asynccnt 0x0" ::: "memory");
    __syncthreads();

    const float* sWself  = &sW[0];
    const float* sWneigh = &sW[4096];
    const float* sBias   = &sW[8192];

    const int lane = tid & 31;
    const int wave = tid >> 5;
    const int tile = blockIdx.x * (GB_THREADS >> 5) + wave;  // 16-row tile
    const int r0 = tile * 16;
    if (r0 >= n_nodes) return;  // uniform per wave; AFTER the barrier

    const int lmod = lane & 15;
    const int lhi  = lane >> 4;                              // 0 or 1
    const int row  = r0 + lmod;
    const int rowc = (row < n_nodes) ? row : (n_nodes - 1);  // clamp for loads

    const float invd = 1.0f / fmaxf(DEG[rowc], 1.0f);
    const float* xrow = X   + (size_t)rowc * 64;
    const float* nrow = NBR + (size_t)rowc * 64;

    for (int nt = 0; nt < 4; ++nt) {
        const int col = nt * 16 + lmod;
        v8f acc = {};

        // self term: acc += X_tile @ Wself_tile
        #pragma unroll
        for (int kk = 0; kk < 16; ++kk) {
            const int kbase = kk * 4 + lhi * 2;
            v2f a, b;
            a.x = xrow[kbase];
            a.y = xrow[kbase + 1];
            b.x = sWself[kbase * 64 + col];
            b.y = sWself[(kbase + 1) * 64 + col];
            acc = __builtin_amdgcn_wmma_f32_16x16x4_f32(
                false, a, false, b, (short)0, acc, false, false);
        }
        // neighbor term: acc += (NBR_tile * invd) @ Wneigh_tile
        #pragma unroll
        for (int kk = 0; kk < 16; ++kk) {
            const int kbase = kk * 4 + lhi * 2;
            v2f a, b;
            a.x = nrow[kbase] * invd;
            a.y = nrow[kbase + 1] * invd;
            b.x = sWneigh[kbase * 64 + col];
            b.y = sWneigh[(kbase + 1) * 64 + col];
            acc = __builtin_amdgcn_wmma_f32_16x16x4_f32(
                false, a, false, b, (short)0, acc, false, false);
        }

        const float bv = sBias[col];
        #pragma unroll
        for (int g = 0; g < 8; ++g) {
            int orow = r0 + g + lhi * 8;
            float v = acc[g] + bv;
            if (do_relu) v = fmaxf(v, 0.0f);
            if (orow < n_nodes)
                OUT[(size_t)orow * 64 + col] = v;
        }
    }
}

// ---------------------------------------------------------------------------
// Launcher
// ---------------------------------------------------------------------------
extern "C" void kernel_launch(void* const* d_in, const int* in_sizes, int n_in,
                              void* d_out, int out_size, void* d_ws, size_t ws_size,
                              hipStream_t stream) {
    const float* feats    = (const float*)d_in[0];
    const float* W_self1  = (const float*)d_in[1];
    const float* W_neigh1 = (const float*)d_in[2];
    const float* b1       = (const float*)d_in[3];
    const float* W_self2  = (const float*)d_in[4];
    const float* W_neigh2 = (const float*)d_in[5];
    const float* b2       = (const float*)d_in[6];
    const int*   esrc     = (const int*)d_in[7];
    const int*   edst     = (const int*)d_in[8];

    const int N = in_sizes[0] / 64;
    const long long E = in_sizes[7];

    // workspace: [ NBR: N*64 | DEG: N | H1: N*64 ] floats
    float* nbr = (float*)d_ws;
    float* deg = nbr + (size_t)N * 64;
    float* h1  = deg + (size_t)N;

    float* out = (float*)d_out;

    const long long nz = (long long)N * 65;             // NBR + DEG contiguous
    const long long aitems = E * 16;                    // float4 granularity
    const int ZB = 256, AB = 256;
    const int zgrid = (int)((nz + ZB - 1) / ZB);
    const int agrid = (int)((aitems + AB - 1) / AB);
    const int tiles = (N + 15) / 16;
    const int ggrid = (tiles + (GB_THREADS / 32) - 1) / (GB_THREADS / 32);

    // ---- Layer 1 ----
    sage_zero_f32<<<zgrid, ZB, 0, stream>>>(nbr, nz);
    sage_aggregate<<<agrid, AB, 0, stream>>>((const float4*)feats, esrc, edst,
                                             nbr, deg, aitems);
    sage_gemm_wmma<<<ggrid, GB_THREADS, 0, stream>>>(feats, nbr, deg,
                                                     W_self1, W_neigh1, b1,
                                                     h1, N, /*relu=*/1);
    // ---- Layer 2 ----
    sage_zero_f32<<<zgrid, ZB, 0, stream>>>(nbr, nz);
    sage_aggregate<<<agrid, AB, 0, stream>>>((const float4*)h1, esrc, edst,
                                             nbr, deg, aitems);
    sage_gemm_wmma<<<ggrid, GB_THREADS, 0, stream>>>(h1, nbr, deg,
                                                     W_self2, W_neigh2, b2,
                                                     out, N, /*relu=*/0);
}